// PointToDense_25383256720087
// MI455X (gfx1250) — compile-verified
//
#include <hip/hip_runtime.h>
#include <float.h>

typedef __attribute__((ext_vector_type(2))) float v2f;
typedef __attribute__((ext_vector_type(4))) float v4f;
typedef __attribute__((ext_vector_type(8))) float v8f;

#define BATCH 16
#define NPTS  2048
#define KNN   10
#define C1    64      // h1 channels
#define C2    256     // output channels
#define KDIM  128     // 2*C1

// ---------------------------------------------------------------------------
// Kernel 1: KNN (k=10) per point. One thread per (b,n). Points + |x|^2 staged
// in LDS; wave-uniform m loop -> broadcast LDS reads. Register insertion list
// with strict-< compare reproduces top_k's lower-index tie-break.
// ---------------------------------------------------------------------------
__global__ __launch_bounds__(256) void knn_kernel(const float* __restrict__ xy,
                                                  int* __restrict__ idx)
{
    __shared__ float sx[NPTS];
    __shared__ float sy[NPTS];
    __shared__ float sq[NPTS];
    const int b = blockIdx.x >> 3;
    const int n = ((blockIdx.x & 7) << 8) + threadIdx.x;
    const float* base = xy + (size_t)b * NPTS * 2;

    for (int i = threadIdx.x; i < NPTS; i += 256) {
        float x = base[i * 2 + 0];
        float y = base[i * 2 + 1];
        sx[i] = x; sy[i] = y; sq[i] = x * x + y * y;
    }
    __syncthreads();

    const float cx = sx[n], cy = sy[n], sqn = sq[n];
    float bd[KNN]; int bi[KNN];
#pragma unroll
    for (int j = 0; j < KNN; ++j) { bd[j] = FLT_MAX; bi[j] = 0; }

    for (int m = 0; m < NPTS; ++m) {
        float inner = cx * sx[m] + cy * sy[m];
        float d = sqn - 2.0f * inner + sq[m];
        if (d < bd[KNN - 1]) {
            float v = d; int vi = m;
#pragma unroll
            for (int j = 0; j < KNN; ++j) {
                if (v < bd[j]) {
                    float tb = bd[j]; int ti = bi[j];
                    bd[j] = v;  bi[j] = vi;
                    v = tb;     vi = ti;
                }
            }
        }
    }
    int* op = idx + (size_t)(b * NPTS + n) * KNN;
#pragma unroll
    for (int j = 0; j < KNN; ++j) op[j] = bi[j];
}

// ---------------------------------------------------------------------------
// Kernel 2: EdgeConv1 (4 -> 64), tiny GEMM: plain VALU, W1/b1 in LDS
// (uniform broadcast reads). max over k fused; bias+lrelu applied after max
// (valid since lrelu is monotone and bias is k-invariant).
// ---------------------------------------------------------------------------
__global__ __launch_bounds__(256) void edge1_kernel(const float* __restrict__ xy,
                                                    const int* __restrict__ idx,
                                                    const float* __restrict__ W1,
                                                    const float* __restrict__ b1,
                                                    float* __restrict__ h1)
{
    __shared__ float w1s[C1 * 4];
    __shared__ float b1s[C1];
    if (threadIdx.x < C1 * 4) w1s[threadIdx.x] = W1[threadIdx.x];
    if (threadIdx.x < C1)     b1s[threadIdx.x] = b1[threadIdx.x];
    __syncthreads();

    const int b = blockIdx.x >> 3;
    const int n = ((blockIdx.x & 7) << 8) + threadIdx.x;
    const float* base = xy + (size_t)b * NPTS * 2;
    const float cx = base[n * 2 + 0];
    const float cy = base[n * 2 + 1];
    const int* ip = idx + (size_t)(b * NPTS + n) * KNN;

    float mh[C1];
#pragma unroll
    for (int o = 0; o < C1; ++o) mh[o] = -FLT_MAX;

    for (int k = 0; k < KNN; ++k) {
        int j = ip[k];
        float nx = base[j * 2 + 0];
        float ny = base[j * 2 + 1];
        float e0 = nx - cx, e1 = ny - cy;
#pragma unroll
        for (int o = 0; o < C1; ++o) {
            float h = e0 * w1s[o * 4 + 0] + e1 * w1s[o * 4 + 1]
                    + cx * w1s[o * 4 + 2] + cy * w1s[o * 4 + 3];
            mh[o] = fmaxf(mh[o], h);
        }
    }
    float* hp = h1 + (size_t)(b * NPTS + n) * C1;
#pragma unroll
    for (int o = 0; o < C1; ++o) {
        float h = mh[o] + b1s[o];
        hp[o] = (h > 0.0f) ? h : 0.2f * h;
    }
}

// ---------------------------------------------------------------------------
// Kernel 3: EdgeConv2 (128 -> 256) on the matrix pipes, spill-free layout.
//
// Per wave: M-tile = 16 consecutive n at fixed neighbor k; loop N-tiles (16)
// OUTERMOST so only one tile's B-fragments (64 VGPRs) are live across the
// k-loop -- LICM keeps them in registers instead of spilling to scratch.
//
// Split GEMM:  h = (neigh - ctr) . W2[:, 0:64]  +  ctr . W2[:, 64:128].
// The ctr half (accC) is k-invariant, so it is computed once per tile (16
// WMMAs) and only the neighbor half runs per k (16 WMMAs, as two independent
// 8-deep chains a0/a1 to cut the serial acc latency). max_k commutes with the
// k-invariant addend and leaky-relu is monotone, so:
//   out = lrelu( max_k(neigh_k) + accC + bias ).
//
// A layout (ISA 16x4 f32): lane L holds row M=L%16, K = {0,1}+2*(L>=16).
// B layout: lane L holds N=L%16, same K split.  C/D: vgpr v -> M = v+8*(L>=16).
// ---------------------------------------------------------------------------
__global__ __launch_bounds__(128) void edge2_wmma_kernel(const float* __restrict__ h1,
                                                         const int* __restrict__ idx,
                                                         const float* __restrict__ W2,
                                                         const float* __restrict__ b2,
                                                         float* __restrict__ out)
{
    const int wave  = (blockIdx.x << 2) + (threadIdx.x >> 5);
    const int lane  = threadIdx.x & 31;
    const int b     = wave >> 7;          // 128 chunks of 16 points per batch
    const int chunk = wave & 127;
    const int n_base = chunk << 4;
    const int mlane  = lane & 15;
    const int hl     = lane >> 4;         // 0 or 1 (which half-wave)
    const int off    = hl << 1;           // K sub-offset {0,1} vs {2,3}

    const int n = n_base + mlane;
    const float* hn = h1 + (size_t)(b * NPTS + n) * C1;
    const int*   ip = idx + (size_t)(b * NPTS + n) * KNN;

    // Neighbor indices: k-, nt-invariant, keep in registers.
    int jidx[KNN];
#pragma unroll
    for (int k = 0; k < KNN; ++k) jidx[k] = ip[k];

    // Center-point A fragments (ctr half of e, and subtrahend for neigh half).
    v2f aC[16];
#pragma unroll
    for (int ks = 0; ks < 16; ++ks)
        aC[ks] = *(const v2f*)(hn + 4 * ks + off);

#pragma unroll 1
    for (int nt = 0; nt < 16; ++nt) {
        const int o = nt * 16 + mlane;
        const float* wrow = W2 + (size_t)o * KDIM + off;

        // B fragments for this N-tile: neigh half (c<64) + ctr half (c>=64).
        v2f Bn[16], Bc[16];
#pragma unroll
        for (int ks = 0; ks < 16; ++ks) Bn[ks] = *(const v2f*)(wrow + 4 * ks);
#pragma unroll
        for (int ks = 0; ks < 16; ++ks) Bc[ks] = *(const v2f*)(wrow + 64 + 4 * ks);

        // k-invariant contribution: ctr . W2[:, 64:128]
        v8f accC = {0.f, 0.f, 0.f, 0.f, 0.f, 0.f, 0.f, 0.f};
#pragma unroll
        for (int ks = 0; ks < 16; ++ks)
            accC = __builtin_amdgcn_wmma_f32_16x16x4_f32(
                     false, aC[ks], false, Bc[ks], (short)0, accC, false, false);

        v8f maxv;
#pragma unroll
        for (int v = 0; v < 8; ++v) maxv[v] = -FLT_MAX;

#pragma unroll 1
        for (int k = 0; k < KNN; ++k) {
            const float* hj = h1 + (size_t)(b * NPTS + jidx[k]) * C1;
            v2f aF[16];
#pragma unroll
            for (int ks = 0; ks < 16; ++ks)
                aF[ks] = *(const v2f*)(hj + 4 * ks + off) - aC[ks];

            v8f a0 = {0.f, 0.f, 0.f, 0.f, 0.f, 0.f, 0.f, 0.f};
            v8f a1 = {0.f, 0.f, 0.f, 0.f, 0.f, 0.f, 0.f, 0.f};
#pragma unroll
            for (int ks = 0; ks < 8; ++ks) {
                a0 = __builtin_amdgcn_wmma_f32_16x16x4_f32(
                       false, aF[2 * ks + 0], false, Bn[2 * ks + 0], (short)0, a0, false, false);
                a1 = __builtin_amdgcn_wmma_f32_16x16x4_f32(
                       false, aF[2 * ks + 1], false, Bn[2 * ks + 1], (short)0, a1, false, false);
            }
#pragma unroll
            for (int v = 0; v < 8; ++v)
                maxv[v] = fmaxf(maxv[v], a0[v] + a1[v]);
        }

        // Epilogue for this N-tile: + ctr part + bias, leaky-relu, store
        // transposed out[b, o, n] (two contiguous b128 stores per lane).
        const float bias = b2[o];
        float r[8];
#pragma unroll
        for (int v = 0; v < 8; ++v) {
            float h = maxv[v] + accC[v] + bias;
            r[v] = (h > 0.0f) ? h : 0.2f * h;
        }
        float* op = out + (size_t)b * C2 * NPTS + (size_t)o * NPTS
                        + n_base + (hl << 3);
        v4f r0 = {r[0], r[1], r[2], r[3]};
        v4f r1 = {r[4], r[5], r[6], r[7]};
        *(v4f*)(op)     = r0;
        *(v4f*)(op + 4) = r1;
    }
}

// ---------------------------------------------------------------------------
extern "C" void kernel_launch(void* const* d_in, const int* in_sizes, int n_in,
                              void* d_out, int out_size, void* d_ws, size_t ws_size,
                              hipStream_t stream)
{
    const float* xy = (const float*)d_in[0];
    const float* W1 = (const float*)d_in[1];
    const float* b1 = (const float*)d_in[2];
    const float* W2 = (const float*)d_in[3];
    const float* b2 = (const float*)d_in[4];
    float* out = (float*)d_out;

    // Workspace layout: idx (16*2048*10 int = 1.25 MB, 16B aligned) then h1
    // (16*2048*64 f32 = 8 MB). Total < 10 MB.
    int*   idxbuf = (int*)d_ws;
    float* h1     = (float*)((char*)d_ws + (size_t)BATCH * NPTS * KNN * sizeof(int));

    knn_kernel  <<<dim3(BATCH * 8), dim3(256), 0, stream>>>(xy, idxbuf);
    edge1_kernel<<<dim3(BATCH * 8), dim3(256), 0, stream>>>(xy, idxbuf, W1, b1, h1);
    edge2_wmma_kernel<<<dim3(512), dim3(128), 0, stream>>>(h1, idxbuf, W2, b2, out);
}